// GaussionDecomposeBlock_20452634263738
// MI455X (gfx1250) — compile-verified
//
#include <hip/hip_runtime.h>
#include <stdint.h>

// x: (4,1,8,256,256) f32 -> 32 images 256x256; 8-level Gaussian pyramid chain.
// out = [ low:(4,8,8,256,256) | hig:(4,8,8,256,256) ] flat f32.

#define TILE_H 16
#define HALO   16                    // 8 levels * 2 rows/level
#define BUF_H  (TILE_H + 2*HALO)     // 48 rows resident in LDS
#define IMG_W  256
#define IMG_H  256
#define IMG_PX (IMG_W*IMG_H)
#define OUT_HALF 16777216            // 4*8*8*256*256

// 1D Gaussian sigma=1 taps [-2..2], normalized (2D kernel = outer product).
#define GW0 0.05448868454964294f
#define GW1 0.24420134200323348f
#define GW2 0.40261994689424746f

typedef float        v4f __attribute__((ext_vector_type(4)));
typedef unsigned int v4u __attribute__((ext_vector_type(4)));
typedef int          v4i __attribute__((ext_vector_type(4)));
typedef int          v8i __attribute__((ext_vector_type(8)));

// Generic LDS pointer: low 32 bits are the wave-relative LDS byte address (ISA 10.2).
__device__ __forceinline__ uint32_t lds_addr(const void* p) {
  return (uint32_t)(uintptr_t)p;
}

// ---- CDNA5 Tensor Data Mover: 2D tile (256 cols x nrows f32, row stride 256) -> LDS.
// D# built per ISA 8.3/8.4: group0 = {count|flags, lds_addr, global_addr, type=2},
// group1 = {mask/data_size, dims, tile dims, strides}; groups 2/3 zero (dims >2 unused).
__device__ __forceinline__ void tdm_load_2d(uint32_t lds_byte, const float* gaddr, int nrows) {
#if defined(__HIP_DEVICE_COMPILE__) && __has_builtin(__builtin_amdgcn_tensor_load_to_lds)
  uint64_t ga = (uint64_t)(uintptr_t)gaddr;
  v4u g0;
  g0.x = 0x1u;                                     // count=1, user mode
  g0.y = lds_byte;                                 // lds_addr (bytes)
  g0.z = (uint32_t)(ga & 0xffffffffu);             // global_addr[31:0]
  g0.w = (uint32_t)((ga >> 32) & 0x01ffffffu) | (2u << 30);  // addr[56:32] | type=2
  const uint32_t td0 = 256u, td1 = 256u;           // tensor dims (elements)
  const uint32_t tl0 = 256u, tl1 = (uint32_t)nrows;// tile dims
  const uint64_t s0 = 256u, s1 = 65536u;           // strides (elements)
  v8i g1;
  g1[0] = (int)(2u << 16);                               // data_size=4B; no mask/pad/iter
  g1[1] = (int)((td0 & 0xffffu) << 16);                  // tensor_dim0[15:0]
  g1[2] = (int)((td0 >> 16) | ((td1 & 0xffffu) << 16));  // td0 hi | td1 lo
  g1[3] = (int)((td1 >> 16) | (tl0 << 16));              // td1 hi | tile_dim0
  g1[4] = (int)(tl1 & 0xffffu);                          // tile_dim1 | tile_dim2=0
  g1[5] = (int)(uint32_t)(s0 & 0xffffffffu);             // dim0_stride[31:0]
  g1[6] = (int)((uint32_t)(s0 >> 32) | ((uint32_t)(s1 & 0xffffu) << 16));
  g1[7] = (int)(uint32_t)((s1 >> 16) & 0xffffffffu);     // dim1_stride[47:16]
  v4i g2 = {0, 0, 0, 0};
  v4i g3 = {0, 0, 0, 0};
#if __has_include(<hip/amd_detail/amd_gfx1250_TDM.h>)
  v8i g4 = {0, 0, 0, 0, 0, 0, 0, 0};               // therock toolchain: 6-arg form
  __builtin_amdgcn_tensor_load_to_lds(g0, g1, g2, g3, g4, 0);
#else
  __builtin_amdgcn_tensor_load_to_lds(g0, g1, g2, g3, 0); // ROCm 7.2: 5-arg form
#endif
#else
  // Fallback: per-lane async global->LDS b128 copies (ASYNCcnt path).
  const int tid = threadIdx.x;
  const int c4 = (tid & 63) << 2, rg = tid >> 6;
  for (int br = rg; br < nrows; br += 4) {
    asm volatile("global_load_async_to_lds_b128 %0, %1, off"
                 :: "v"(lds_byte + (uint32_t)(br * IMG_W + c4) * 4u),
                    "v"(gaddr + br * IMG_W + c4) : "memory");
  }
#endif
}

__device__ __forceinline__ void tdm_wait0() {
#if defined(__HIP_DEVICE_COMPILE__)
#if __has_builtin(__builtin_amdgcn_tensor_load_to_lds)
#if __has_builtin(__builtin_amdgcn_s_wait_tensorcnt)
  __builtin_amdgcn_s_wait_tensorcnt(0);
#else
  asm volatile("s_wait_tensorcnt 0" ::: "memory");
#endif
#else
  asm volatile("s_wait_asynccnt 0" ::: "memory");
#endif
#endif
}

extern "C" __global__ __launch_bounds__(256)
void gauss_pyramid_kernel(const float* __restrict__ x, float* __restrict__ out) {
  extern __shared__ float smem[];
  float* IN   = smem;                         // BUF_H * IMG_W
  float* TMP  = smem + BUF_H * IMG_W;         // BUF_H * IMG_W
  float* ORIG = smem + 2 * BUF_H * IMG_W;     // TILE_H * IMG_W

  const int img  = blockIdx.x >> 4;           // 0..31  (= b*8 + n)
  const int tile = blockIdx.x & 15;           // 0..15
  const int r0   = tile * TILE_H;
  const int rB   = r0 - HALO;                 // global row of buffer row 0
  const int tid  = threadIdx.x;               // 0..255
  const int bi   = img >> 3;
  const int ni   = img & 7;
  const float* src = x + (size_t)img * IMG_PX;

  const v4f zero4 = {0.f, 0.f, 0.f, 0.f};
  const int cb = tid & 63;          // float4 column block 0..63
  const int c4 = cb << 2;           // column base (16B aligned)
  const int rg = tid >> 6;          // row group 0..3

  // ---- Stage tile (+halo): TDM DMA for the in-image row range (wave 0 issues;
  // EXEC-ignored single instruction per descriptor), ds zero-fill for halo rows
  // outside the image (exact zero-pad semantics).
  const int gstart = (rB < 0) ? 0 : rB;
  const int gend   = (rB + BUF_H > IMG_H) ? IMG_H : rB + BUF_H;
  if (tid < 32) {
    tdm_load_2d(lds_addr(&IN[(gstart - rB) * IMG_W]), src + gstart * IMG_W, gend - gstart);
    tdm_load_2d(lds_addr(&ORIG[0]), src + r0 * IMG_W, TILE_H);
    tdm_wait0();
  }
  #pragma unroll
  for (int j = 0; j < BUF_H / 4; ++j) {
    int br = rg + j * 4;
    int gr = rB + br;
    if ((unsigned)gr >= (unsigned)IMG_H)
      *(v4f*)(&IN[br * IMG_W + c4]) = zero4;
  }
  __syncthreads();

  const bool hasL = (cb > 0);
  const bool hasR = (cb < 63);

  // ---- 8-level LDS-resident pyramid chain (separable 5-tap, zero-padded).
  #pragma unroll
  for (int lev = 0; lev < 8; ++lev) {
    const int rlo = 2 * lev;                  // valid input rows [rlo, rhi)
    const int rhi = BUF_H - 2 * lev;

    // Row pass: IN -> TMP. Per thread: 3 aligned b128 reads -> 4 outputs.
    for (int br = rlo + rg; br < rhi; br += 4) {
      const v4f* rowv = (const v4f*)&IN[br * IMG_W];
      v4f L = hasL ? rowv[cb - 1] : zero4;    // cols c4-4..c4-1
      v4f C = rowv[cb];                        // cols c4  ..c4+3
      v4f R = hasR ? rowv[cb + 1] : zero4;    // cols c4+4..c4+7
      v4f o;
      o.x = GW0 * (L.z + C.z) + GW1 * (L.w + C.y) + GW2 * C.x;  // x = c4
      o.y = GW0 * (L.w + C.w) + GW1 * (C.x + C.z) + GW2 * C.y;  // x = c4+1
      o.z = GW0 * (C.x + R.x) + GW1 * (C.y + C.w) + GW2 * C.z;  // x = c4+2
      o.w = GW0 * (C.y + R.y) + GW1 * (C.z + R.x) + GW2 * C.w;  // x = c4+3
      *(v4f*)&TMP[br * IMG_W + c4] = o;
    }
    __syncthreads();

    // Col pass: TMP -> IN; out-of-image rows forced to 0; tile rows stream out
    // as non-temporal b128 stores (write-once 128 MB stream, don't thrash L2).
    for (int br = rlo + 2 + rg; br < rhi - 2; br += 4) {
      const float* tc = &TMP[br * IMG_W + c4];
      v4f t0 = *(const v4f*)(tc - 2 * IMG_W);
      v4f t1 = *(const v4f*)(tc - 1 * IMG_W);
      v4f t2 = *(const v4f*)(tc);
      v4f t3 = *(const v4f*)(tc + 1 * IMG_W);
      v4f t4 = *(const v4f*)(tc + 2 * IMG_W);
      v4f v = GW0 * (t0 + t4) + GW1 * (t1 + t3) + GW2 * t2;
      int gr = rB + br;
      if ((unsigned)gr >= (unsigned)IMG_H) v = zero4;
      *(v4f*)&IN[br * IMG_W + c4] = v;
      if (br >= HALO && br < HALO + TILE_H) {
        // low[b][lev][n][gr][x] ; hig = orig - low
        int off = ((bi * 8 + lev) * 8 + ni) * IMG_PX + gr * IMG_W + c4;
        v4f o = *(const v4f*)&ORIG[(br - HALO) * IMG_W + c4];
        __builtin_nontemporal_store(v,     (v4f*)&out[off]);
        __builtin_nontemporal_store(o - v, (v4f*)&out[off + OUT_HALF]);
      }
    }
    __syncthreads();
  }
}

extern "C" void kernel_launch(void* const* d_in, const int* in_sizes, int n_in,
                              void* d_out, int out_size, void* d_ws, size_t ws_size,
                              hipStream_t stream) {
  (void)in_sizes; (void)n_in; (void)d_ws; (void)ws_size; (void)out_size;
  const float* x = (const float*)d_in[0];   // (4,1,8,256,256) f32; d_in[1]=dimOut(=8) hardcoded
  float* out = (float*)d_out;
  const size_t smem = (size_t)(2 * BUF_H + TILE_H) * IMG_W * sizeof(float); // 112 KB
  dim3 grid(512), block(256);               // 32 images * 16 row-tiles; 8 waves/WG
  gauss_pyramid_kernel<<<grid, block, smem, stream>>>(x, out);
}